// GCNLayer_27779848471367
// MI455X (gfx1250) — compile-verified
//
#include <hip/hip_runtime.h>

#define IN_F   128
#define OUT_F  128
#define M_TILE 64
#define LN_EPS 1e-5f

typedef __attribute__((ext_vector_type(2))) float v2f;
typedef __attribute__((ext_vector_type(8))) float v8f;

// ---------------------------------------------------------------------------
// Kernel 1: zero the accumulator (d_out doubles as the h scatter buffer)
// ---------------------------------------------------------------------------
__global__ void gcn_zero_kernel(float4* __restrict__ out, int n4) {
    int i = blockIdx.x * blockDim.x + threadIdx.x;
    if (i < n4) {
        float4 z; z.x = 0.f; z.y = 0.f; z.z = 0.f; z.w = 0.f;
        out[i] = z;
    }
}

// ---------------------------------------------------------------------------
// Kernel 2: edge scatter-add. One wave (32 lanes) per edge; each lane moves
// 4 contiguous floats (float4 gather + 4 global f32 atomics). src/dst are
// wave-uniform -> scalar loads. All traffic is L2-resident (51MB tables).
// ---------------------------------------------------------------------------
__global__ void gcn_scatter_kernel(const float* __restrict__ feat,
                                   const int*   __restrict__ src,
                                   const int*   __restrict__ dst,
                                   float*       __restrict__ h,
                                   int n_edges) {
    int gid  = blockIdx.x * blockDim.x + threadIdx.x;
    int e    = gid >> 5;
    int lane = gid & 31;
    if (e >= n_edges) return;
    int s = src[e];
    int d = dst[e];
    const float4 v = *(const float4*)(feat + (size_t)s * IN_F + lane * 4);
    float* p = h + (size_t)d * IN_F + lane * 4;
    atomicAdd(p + 0, v.x);
    atomicAdd(p + 1, v.y);
    atomicAdd(p + 2, v.z);
    atomicAdd(p + 3, v.w);
}

// ---------------------------------------------------------------------------
// Kernel 3: fused LayerNorm + GEMM (h @ W^T + b) using V_WMMA_F32_16X16X4_F32.
// 64 nodes / block, 256 threads = 8 waves. Wave w owns output columns
// [16w, 16w+16) for all 4 M-subtiles; B fragment reused across M.
// In-place on `out`: rows are fully read into LDS before being overwritten.
// ---------------------------------------------------------------------------
__global__ __launch_bounds__(256) void gcn_ln_gemm_kernel(
    float*       __restrict__ out,    // in: scatter result h, out: final
    const float* __restrict__ W,      // [OUT_F][IN_F]
    const float* __restrict__ bias,   // [OUT_F]
    const float* __restrict__ gamma,  // [IN_F]
    const float* __restrict__ beta,   // [IN_F]
    int n_nodes) {

    __shared__ float shW[IN_F * OUT_F];     // 64 KB, shW[k][n] = W[n][k]
    __shared__ float shH[M_TILE * IN_F];    // 32 KB
    __shared__ float red_s[M_TILE * 4];
    __shared__ float red_q[M_TILE * 4];
    __shared__ float s_mu[M_TILE];
    __shared__ float s_rs[M_TILE];

    const int tid   = threadIdx.x;
    const int node0 = blockIdx.x * M_TILE;

    // Stage W transposed into LDS: shW[i*OUT_F + o] = W[o][i]
    for (int idx = tid; idx < IN_F * OUT_F; idx += 256) {
        int o = idx / IN_F;
        int i = idx - o * IN_F;
        shW[i * OUT_F + o] = W[idx];
    }
    // Stage the 64 h rows (float4 loads, zero-pad past n_nodes)
    for (int idx = tid; idx < M_TILE * (IN_F / 4); idx += 256) {
        int row = idx / (IN_F / 4);
        int c4  = idx - row * (IN_F / 4);
        float4 v; v.x = 0.f; v.y = 0.f; v.z = 0.f; v.w = 0.f;
        int g = node0 + row;
        if (g < n_nodes) v = *(const float4*)(out + (size_t)g * IN_F + c4 * 4);
        *(float4*)(shH + row * IN_F + c4 * 4) = v;
    }
    __syncthreads();

    // LayerNorm: 4 threads per node, 32 elements each
    {
        int node = tid >> 2;
        int seg  = tid & 3;
        const float* p = shH + node * IN_F + seg * 32;
        float s = 0.f, q = 0.f;
        #pragma unroll
        for (int j = 0; j < 32; ++j) { float v = p[j]; s += v; q += v * v; }
        red_s[tid] = s;
        red_q[tid] = q;
    }
    __syncthreads();
    if (tid < M_TILE) {
        float s = red_s[tid*4] + red_s[tid*4+1] + red_s[tid*4+2] + red_s[tid*4+3];
        float q = red_q[tid*4] + red_q[tid*4+1] + red_q[tid*4+2] + red_q[tid*4+3];
        float mu  = s * (1.0f / IN_F);
        float var = q * (1.0f / IN_F) - mu * mu;
        s_mu[tid] = mu;
        s_rs[tid] = rsqrtf(var + LN_EPS);
    }
    __syncthreads();
    {
        int node = tid >> 2;
        int seg  = tid & 3;
        float mu = s_mu[node];
        float rs = s_rs[node];
        float* p = shH + node * IN_F + seg * 32;
        const float* gp = gamma + seg * 32;
        const float* bp = beta  + seg * 32;
        #pragma unroll
        for (int j = 0; j < 32; ++j)
            p[j] = (p[j] - mu) * rs * gp[j] + bp[j];
    }
    __syncthreads();

    // ---- WMMA phase: D = A(16x4) x B(4x16) + C, fp32 exact ----
    const int wave  = tid >> 5;
    const int lane  = tid & 31;
    const int laneM = lane & 15;
    const int hi    = lane >> 4;      // lanes 16-31 hold K+2 of the fragment
    const int n0    = wave * 16;

    v8f acc[4];
    acc[0] = (v8f){}; acc[1] = (v8f){}; acc[2] = (v8f){}; acc[3] = (v8f){};

    #pragma unroll 8
    for (int k = 0; k < IN_F; k += 4) {
        const int k0 = k + hi * 2;
        // B fragment: rows k..k+3 of W^T, column n0+laneM
        v2f bf;
        bf.x = shW[(k0 + 0) * OUT_F + n0 + laneM];
        bf.y = shW[(k0 + 1) * OUT_F + n0 + laneM];
        #pragma unroll
        for (int mt = 0; mt < 4; ++mt) {
            // A fragment: row (mt*16 + laneM), cols k0, k0+1
            const float* hp = shH + (mt * 16 + laneM) * IN_F + k0;
            v2f a;
            a.x = hp[0];
            a.y = hp[1];
            acc[mt] = __builtin_amdgcn_wmma_f32_16x16x4_f32(
                /*neg_a=*/false, a, /*neg_b=*/false, bf,
                /*c_mod=*/(short)0, acc[mt],
                /*reuse_a=*/false, /*reuse_b=*/false);
        }
    }

    // Store: C/D layout -> VGPR j holds M=j (lanes 0-15) / M=j+8 (lanes 16-31)
    const float bn = bias[n0 + laneM];
    #pragma unroll
    for (int mt = 0; mt < 4; ++mt) {
        #pragma unroll
        for (int j = 0; j < 8; ++j) {
            int m = mt * 16 + hi * 8 + j;
            int g = node0 + m;
            if (g < n_nodes)
                out[(size_t)g * OUT_F + n0 + laneM] = acc[mt][j] + bn;
        }
    }
}

// ---------------------------------------------------------------------------
extern "C" void kernel_launch(void* const* d_in, const int* in_sizes, int n_in,
                              void* d_out, int out_size, void* d_ws, size_t ws_size,
                              hipStream_t stream) {
    const float* feat  = (const float*)d_in[0];
    const int*   src   = (const int*)  d_in[1];
    const int*   dst   = (const int*)  d_in[2];
    const float* W     = (const float*)d_in[3];
    const float* bias  = (const float*)d_in[4];
    const float* gamma = (const float*)d_in[5];
    const float* beta  = (const float*)d_in[6];
    float*       out   = (float*)d_out;

    const int n_nodes = in_sizes[0] / IN_F;
    const int n_edges = in_sizes[1];

    // 1) zero the accumulator (d_out is poisoned by harness; we own init)
    const int n4 = out_size / 4;
    gcn_zero_kernel<<<(n4 + 255) / 256, 256, 0, stream>>>((float4*)out, n4);

    // 2) edge scatter-add: one wave per edge
    const long long sthreads = (long long)n_edges * 32;
    const int sblocks = (int)((sthreads + 255) / 256);
    gcn_scatter_kernel<<<sblocks, 256, 0, stream>>>(feat, src, dst, out, n_edges);

    // 3) fused LayerNorm + WMMA GEMM (+bias), in-place on out
    const int mblocks = (n_nodes + M_TILE - 1) / M_TILE;
    gcn_ln_gemm_kernel<<<mblocks, 256, 0, stream>>>(out, W, bias, gamma, beta, n_nodes);
}